// cropping_patch_7576322310348
// MI455X (gfx1250) — compile-verified
//
#include <hip/hip_runtime.h>
#include <cstdint>
#include <cstddef>

// ---------------------------------------------------------------------------
// Cropping-patch gather for MI455X (gfx1250, wave32).
// Pure data movement (~0.72 GB HBM traffic, ~31 us floor @ 23.3 TB/s).
// Data path: global --(async DMA)--> LDS --(async DMA)--> global, so the
// payload never occupies VGPRs; ASYNCcnt + s_wait_asynccnt order the DMAs.
// ---------------------------------------------------------------------------

#if defined(__HIP_DEVICE_COMPILE__)
#if __has_builtin(__builtin_amdgcn_global_load_async_to_lds_b32) && \
    __has_builtin(__builtin_amdgcn_s_wait_asynccnt)
#define CROP_ASYNC_LOAD 1
#endif
#if __has_builtin(__builtin_amdgcn_global_store_async_from_lds_b32) && \
    __has_builtin(__builtin_amdgcn_s_wait_asynccnt)
#define CROP_ASYNC_STORE 1
#endif
#endif

#if defined(__HIP_DEVICE_COMPILE__)
// Builtin signature (from hipcc diagnostic): addrspace(1) int* global side,
// addrspace(3) int* LDS side, plus two integer immediates (offset, cpol).
typedef __attribute__((address_space(1))) int as1_int;
typedef __attribute__((address_space(3))) int as3_int;
#endif

namespace {
constexpr int T_ = 256;
constexpr int C_ = 256;
constexpr int H_ = 56;
constexpr int W_ = 56;
}

template <int SIZE, int HALF, int KP>
__global__ __launch_bounds__(SIZE * SIZE)
void crop_kernel(const float* __restrict__ src,
                 const float* __restrict__ row,
                 const float* __restrict__ col,
                 float* __restrict__ out)
{
    __shared__ float smem[SIZE * SIZE];

    const int ch  = blockIdx.x;          // channel   [0,256)
    const int t   = blockIdx.y;          // frame     [0,256)
    const int tid = threadIdx.x;         // [0, SIZE*SIZE)
    const int y   = tid / SIZE;          // constant divisor -> mul/shift
    const int x   = tid - y * SIZE;

    // Wave-uniform crop origin. float truncation == jnp astype(int32) for
    // positive inputs; row/col are uniform[0,1) so always positive.
    const float rv = row[t * 7 + KP];
    const float cv = col[t * 7 + KP];
    const int r0 = (int)(rv * (float)H_) - HALF;
    const int c0 = (int)(cv * (float)W_) - HALF;

    const int sy = r0 + y;
    const int sx = c0 + x;
    const bool valid = ((unsigned)sy < (unsigned)H_) &&
                       ((unsigned)sx < (unsigned)W_);

    const size_t plane   = (size_t)(t * C_ + ch) * (size_t)(H_ * W_);
    const float* gp      = src + plane + (size_t)sy * W_ + sx;   // only deref'd if valid
    const size_t out_off = (size_t)(t * C_ + ch) * (size_t)(SIZE * SIZE) + tid;

#if CROP_ASYNC_LOAD
    if (valid) {
        // DMA one dword per active lane straight into this lane's LDS slot.
        __builtin_amdgcn_global_load_async_to_lds_b32(
            (as1_int*)gp, (as3_int*)&smem[tid], /*offset=*/0, /*cpol=*/0);
    } else {
        smem[tid] = 0.0f;                // zero-pad region
    }
    __builtin_amdgcn_s_wait_asynccnt(0); // async LDS writes visible
    __syncthreads();                     // drains DScnt for the zero stores
#else
    smem[tid] = valid ? *gp : 0.0f;
    __syncthreads();
#endif

#if CROP_ASYNC_STORE
    // Stream LDS -> global without a VGPR round trip; writes are linear in
    // the output, i.e. perfectly coalesced. Implicit wait-idle at s_endpgm
    // drains the remaining ASYNCcnt.
    __builtin_amdgcn_global_store_async_from_lds_b32(
        (as1_int*)(out + out_off), (as3_int*)&smem[tid], /*offset=*/0, /*cpol=*/0);
#else
    out[out_off] = smem[tid];
#endif
}

extern "C" void kernel_launch(void* const* d_in, const int* in_sizes, int n_in,
                              void* d_out, int out_size, void* d_ws, size_t ws_size,
                              hipStream_t stream)
{
    (void)in_sizes; (void)n_in; (void)out_size; (void)d_ws; (void)ws_size;

    const float* src = (const float*)d_in[0];   // (256,256,56,56) fp32
    const float* row = (const float*)d_in[1];   // (256,7) fp32
    const float* col = (const float*)d_in[2];   // (256,7) fp32
    float*       out = (float*)d_out;           // face | L | R, flat fp32

    constexpr size_t FACE_ELEMS = (size_t)T_ * C_ * 16 * 16;   // 16,777,216
    constexpr size_t HAND_ELEMS = (size_t)T_ * C_ * 24 * 24;   // 37,748,736

    const dim3 grid(C_, T_);

    // Reference returns (face, L, R) = (kp0, kp2, kp1) concatenated.
    crop_kernel<16,  8, 0><<<grid, 16 * 16, 0, stream>>>(src, row, col, out);
    crop_kernel<24, 12, 2><<<grid, 24 * 24, 0, stream>>>(src, row, col,
                                                         out + FACE_ELEMS);
    crop_kernel<24, 12, 1><<<grid, 24 * 24, 0, stream>>>(src, row, col,
                                                         out + FACE_ELEMS + HAND_ELEMS);
}